// Experts_5669356832625
// MI455X (gfx1250) — compile-verified
//
#include <hip/hip_runtime.h>
#include <hip/hip_bf16.h>

typedef float v2f __attribute__((ext_vector_type(2)));
typedef float v8f __attribute__((ext_vector_type(8)));

#define DIN     256
#define DOUT    256
#define XS_STR  (DIN + 4)     // x-tile LDS row stride (A-reads hit all 64 banks)
#define KCHUNK  32
// Weight panel stored K-pair interleaved: wkp[p][2*n + (k&1)], p = k/2.
// Row stride 544 floats => 544 mod 64 = 32 banks: the two half-waves of a
// WMMA B-read (row pairs p and p+1) land in disjoint 32-bank halves.
#define WKP_STR (2 * DOUT + 32)

// ---------------------------------------------------------------------------
// Kernel 1: deterministic per-expert bucketing (one wave, lane e = expert e).
// Ordered scans -> stable token order -> bitwise-deterministic pipeline.
// ---------------------------------------------------------------------------
__global__ __launch_bounds__(32)
void moe_bucket_kernel(const int* __restrict__ index,
                       int* __restrict__ counts,
                       int* __restrict__ offsets,
                       int* __restrict__ bucket,
                       int ntok, int nexp) {
    __shared__ int soff[64];
    int e = threadIdx.x;
    int c = 0;
    for (int i = 0; i < ntok; ++i) c += (index[i] == e) ? 1 : 0;
    if (e < nexp) counts[e] = c;
    __syncthreads();
    if (e == 0) {
        int run = 0;
        for (int j = 0; j < nexp; ++j) {
            offsets[j] = run;
            soff[j]    = run;
            run += counts[j];
        }
    }
    __syncthreads();
    if (e < nexp) {
        int w = soff[e];
        for (int i = 0; i < ntok; ++i)
            if (index[i] == e) bucket[w++] = i;
    }
}

// ---------------------------------------------------------------------------
// Kernel 2: gathered expert GEMM, 16 tokens x 256 outputs per block.
// 4 waves x 4 WMMA N-tiles each; V_WMMA_F32_16X16X4_F32 (native fp32 path).
// ---------------------------------------------------------------------------
__global__ __launch_bounds__(128)
void moe_expert_gemm(const float* __restrict__ x,
                     const float* __restrict__ weight,   // (E, DIN, DOUT)
                     const float* __restrict__ bias,     // (E, DOUT)
                     const int*   __restrict__ counts,
                     const int*   __restrict__ offsets,
                     const int*   __restrict__ bucket,
                     float* __restrict__ out) {
    __shared__ float xs[16 * XS_STR];
    __shared__ float wkp[(KCHUNK / 2) * WKP_STR];
    __shared__ int   tok[16];

    const int e   = blockIdx.x;
    const int cnt = counts[e];
    const int t   = blockIdx.y;
    if (t * 16 >= cnt) return;                 // uniform exit: EXEC all-1s below
    const int off = offsets[e];
    const int tid = threadIdx.x;

    if (tid < 16) {
        int r = t * 16 + tid;
        tok[tid] = (r < cnt) ? bucket[off + r] : -1;
    }
    __syncthreads();

    // Stage gathered x rows into LDS: thread -> row tid/8, 32-float segment.
    {
        int r  = tid >> 3;
        int c0 = (tid & 7) * 32;
        int tk = tok[r];
        if (tk >= 0) {
            const float4* src = (const float4*)(x + (size_t)tk * DIN + c0);
#pragma unroll
            for (int q = 0; q < 8; ++q)
                *(float4*)&xs[r * XS_STR + c0 + q * 4] = src[q];
        } else {
            float4 z = make_float4(0.f, 0.f, 0.f, 0.f);
#pragma unroll
            for (int q = 0; q < 8; ++q)
                *(float4*)&xs[r * XS_STR + c0 + q * 4] = z;
        }
    }

    const int wave = tid >> 5;
    const int lane = tid & 31;
    const int half = lane >> 4;      // 0: lanes 0-15, 1: lanes 16-31
    const int nl   = lane & 15;
    const int nb0  = wave * 64;      // this wave's 4 N-tiles cover 64 columns

    const v8f vzero = {0.f, 0.f, 0.f, 0.f, 0.f, 0.f, 0.f, 0.f};
    v8f acc[4];
#pragma unroll
    for (int j = 0; j < 4; ++j) acc[j] = vzero;

    const float* we = weight + (size_t)e * DIN * DOUT;

    for (int k0 = 0; k0 < DIN; k0 += KCHUNK) {
        // Prefetch next K-panel (32 KB) while this chunk computes:
        // 128 threads x 256 B spacing -> global_prefetch_b8 covers the panel.
        if (k0 + KCHUNK < DIN)
            __builtin_prefetch(we + (size_t)(k0 + KCHUNK) * DOUT + tid * 64, 0, 3);

        __syncthreads();
        // Cooperative K-panel load, K-pair interleaved:
        // slot s -> row pair p = s/64, 4-col group c4 = s%64.
#pragma unroll
        for (int s = tid; s < (KCHUNK / 2) * (DOUT / 4); s += 128) {
            int p  = s >> 6;
            int c4 = s & 63;
            float4 va = *(const float4*)(we + (size_t)(k0 + 2 * p    ) * DOUT + c4 * 4);
            float4 vb = *(const float4*)(we + (size_t)(k0 + 2 * p + 1) * DOUT + c4 * 4);
            float* d = &wkp[p * WKP_STR + c4 * 8];
            ((float4*)d)[0] = make_float4(va.x, vb.x, va.y, vb.y);
            ((float4*)d)[1] = make_float4(va.z, vb.z, va.w, vb.w);
        }
        __syncthreads();

#pragma unroll
        for (int kk = 0; kk < KCHUNK; kk += 4) {
            // A (16x4 f32): lane holds row M=nl, K = kk + 2*half + {0,1}
            v2f a = *(const v2f*)&xs[nl * XS_STR + k0 + kk + 2 * half];
            const int p0 = (kk >> 1) + half;     // interleaved row-pair index
#pragma unroll
            for (int j = 0; j < 4; ++j) {
                int col = nb0 + j * 16 + nl;
                // B (4x16 f32): both K values adjacent -> single b64 LDS load
                v2f b = *(const v2f*)&wkp[p0 * WKP_STR + col * 2];
                acc[j] = __builtin_amdgcn_wmma_f32_16x16x4_f32(
                    /*neg_a=*/false, a, /*neg_b=*/false, b,
                    /*c_mod=*/(short)0, acc[j],
                    /*reuse_a=*/false, /*reuse_b=*/false);
            }
        }
    }

    // Epilogue: bias add (per column) + scatter rows back via token list.
#pragma unroll
    for (int j = 0; j < 4; ++j) {
        int   col = nb0 + j * 16 + nl;
        float bv  = bias[(size_t)e * DOUT + col];
#pragma unroll
        for (int i = 0; i < 8; ++i) {
            int m  = half * 8 + i;     // C/D layout: M = vgpr + 8*half
            int tk = tok[m];
            if (tk >= 0)
                out[(size_t)tk * DOUT + col] = acc[j][i] + bv;
        }
    }
}

// ---------------------------------------------------------------------------
extern "C" void kernel_launch(void* const* d_in, const int* in_sizes, int n_in,
                              void* d_out, int out_size, void* d_ws, size_t ws_size,
                              hipStream_t stream) {
    const float* x      = (const float*)d_in[0];   // (B, DIN)
    const int*   index  = (const int*)  d_in[1];   // (B,)
    const float* weight = (const float*)d_in[2];   // (E, DIN, DOUT)
    const float* bias   = (const float*)d_in[3];   // (E, DOUT)
    float*       out    = (float*)d_out;

    const int B = in_sizes[0] / DIN;               // 4096
    const int E = in_sizes[3] / DOUT;              // 32

    int* ws      = (int*)d_ws;
    int* counts  = ws;                             // E
    int* offsets = ws + 64;                        // E
    int* bucket  = ws + 128;                       // B

    moe_bucket_kernel<<<1, 32, 0, stream>>>(index, counts, offsets, bucket, B, E);

    dim3 grid(E, (B + 15) / 16);
    moe_expert_gemm<<<grid, 128, 0, stream>>>(x, weight, bias,
                                              counts, offsets, bucket, out);
}